// HardTripletLoss_25804163514701
// MI455X (gfx1250) — compile-verified
//
#include <hip/hip_runtime.h>
#include <hip/hip_bf16.h>
#include <math.h>

typedef __attribute__((ext_vector_type(2))) float v2f;
typedef __attribute__((ext_vector_type(8))) float v8f;

#define B_N 4096
#define K_DIM 512
#define MARGIN 0.5f
#define NEG_INF_BITS ((int)0xFF800000)
#define POS_INF_BITS ((int)0x7F800000)

// ---------------------------------------------------------------------------
// Kernel 1: row squared-norms + init per-row hardest-pos/neg accumulators.
// One wave (32 lanes) per row.
// ---------------------------------------------------------------------------
__global__ __launch_bounds__(32) void norms_init_kernel(
    const float* __restrict__ X, float* __restrict__ norms,
    int* __restrict__ hpBits, int* __restrict__ hnBits) {
  const int row = blockIdx.x;
  const int lane = threadIdx.x;
  const float* p = X + (size_t)row * K_DIM;
  float s = 0.0f;
#pragma unroll
  for (int k = lane; k < K_DIM; k += 32) {
    float v = p[k];
    s += v * v;
  }
  for (int m = 16; m >= 1; m >>= 1) s += __shfl_xor(s, m, 32);
  if (lane == 0) {
    norms[row] = s;
    hpBits[row] = NEG_INF_BITS;   // -inf: max-accumulator identity
    hnBits[row] = POS_INF_BITS;   // +inf: min-accumulator identity
  }
}

// ---------------------------------------------------------------------------
// Kernel 2: WMMA gram-matrix tiles + fused distance/mask/row-reduction.
// Block = 8 waves. Block b owns rows [16b, 16b+16); wave w sweeps column
// tiles j = w, w+8, ..., 248 (16 cols each). Per tile: K loop of 128
// v_wmma_f32_16x16x4_f32 ops accumulating the 16x16 S sub-tile.
//
// f32 WMMA operand layout (ISA 7.12.2):
//   A 16x4 : lane L holds A[M=L%16][K = 2*(L/16) + r] in VGPR r (r=0,1)
//   B 4x16 : lane L holds B[K = 2*(L/16) + r][N=L%16] in VGPR r (mirrored)
//   C 16x16: VGPR r, lanes 0-15 -> (M=r,     N=lane),
//                     lanes 16-31 -> (M=r+8, N=lane-16)
// Since S = X X^T, B[k][n] = X[colBase+n][k]: both A and B loads are the
// same contiguous float2 pattern with different row bases.
// Epilogue is fully branchless (cndmask selects) so EXEC stays all-1s
// throughout the WMMA region.
// ---------------------------------------------------------------------------
__global__ __launch_bounds__(256) void tile_kernel(
    const float* __restrict__ X, const int* __restrict__ labels,
    const float* __restrict__ norms,
    int* __restrict__ hpBits, int* __restrict__ hnBits) {
  const int lane  = threadIdx.x & 31;
  const int wave  = threadIdx.x >> 5;
  const int group = lane >> 4;   // 0 or 1 (half-wave)
  const int nIdx  = lane & 15;
  const int rowBase = blockIdx.x * 16;

  // Per-lane row metadata for the 8 accumulator VGPRs (constant per block).
  float nrow[8];
  int   lrow[8];
#pragma unroll
  for (int r = 0; r < 8; ++r) {
    const int row = rowBase + r + 8 * group;
    nrow[r] = norms[row];
    lrow[r] = labels[row];
  }

  float hp[8], hn[8];
#pragma unroll
  for (int r = 0; r < 8; ++r) { hp[r] = -INFINITY; hn[r] = INFINITY; }

  // A operand base: row = rowBase + (lane&15), k offset = 2*group (8B aligned)
  const float* pa = X + (size_t)(rowBase + nIdx) * K_DIM + 2 * group;

  for (int j = wave; j < B_N / 16; j += 8) {
    const int colBase = j * 16;
    const float* pb = X + (size_t)(colBase + nIdx) * K_DIM + 2 * group;

    v8f c = {0.0f, 0.0f, 0.0f, 0.0f, 0.0f, 0.0f, 0.0f, 0.0f};
#pragma unroll 8
    for (int k0 = 0; k0 < K_DIM; k0 += 4) {
      v2f a = *(const v2f*)(pa + k0);
      v2f b = *(const v2f*)(pb + k0);
      // D = A x B + C   (8 args: neg_a, A, neg_b, B, c_mod, C, reuse_a, reuse_b)
      c = __builtin_amdgcn_wmma_f32_16x16x4_f32(
          false, a, false, b, (short)0, c, false, false);
    }

    // Epilogue: distances + masked row max/min for this 16x16 tile.
    const int   col  = colBase + nIdx;
    const float ncol = norms[col];
    const int   lcol = labels[col];
#pragma unroll
    for (int r = 0; r < 8; ++r) {
      const int row = rowBase + r + 8 * group;
      const float sq = fmaxf(nrow[r] + ncol - 2.0f * c[r], 0.0f);
      // bare v_sqrt_f32 (~1 ulp): avoids the IEEE refinement expansion
      const float d  = __builtin_amdgcn_sqrtf(sq);
      const bool  eq = (lrow[r] == lcol);
      const float dp = (eq && (row != col)) ? d : -INFINITY;  // pos candidate
      const float dn = eq ? INFINITY : d;                     // neg candidate
      hp[r] = fmaxf(hp[r], dp);
      hn[r] = fminf(hn[r], dn);
    }
  }

  // Reduce across the 16 lanes of each half-wave (same row set), then merge
  // into the global per-row accumulators. Distances >= 0, so signed-int
  // atomic max/min on the float bit patterns is order-correct (and the
  // -inf/+inf sentinels compare correctly as ints).
#pragma unroll
  for (int r = 0; r < 8; ++r) {
    float vp = hp[r], vn = hn[r];
    for (int m = 8; m >= 1; m >>= 1) {
      vp = fmaxf(vp, __shfl_xor(vp, m, 32));
      vn = fminf(vn, __shfl_xor(vn, m, 32));
    }
    if (nIdx == 0) {
      const int row = rowBase + r + 8 * group;
      atomicMax(&hpBits[row], __float_as_int(vp));
      atomicMin(&hnBits[row], __float_as_int(vn));
    }
  }
}

// ---------------------------------------------------------------------------
// Kernel 3: final mean of relu(hp - hn + margin) over valid rows.
// ---------------------------------------------------------------------------
__global__ __launch_bounds__(256) void finalize_kernel(
    const int* __restrict__ hpBits, const int* __restrict__ hnBits,
    float* __restrict__ out) {
  __shared__ float ssum[256];
  __shared__ int   scnt[256];
  const int tid = threadIdx.x;
  float sum = 0.0f;
  int   cnt = 0;
  for (int i = tid; i < B_N; i += 256) {
    const int hb = hpBits[i];
    const int nb = hnBits[i];
    // hb == -inf bits  <=> row had no (non-self) positive
    // nb == +inf bits  <=> row had no negative
    if (hb != NEG_INF_BITS && nb != POS_INF_BITS) {
      const float v = __int_as_float(hb) - __int_as_float(nb) + MARGIN;
      sum += (v > 0.0f) ? v : 0.0f;
      cnt += 1;
    }
  }
  ssum[tid] = sum;
  scnt[tid] = cnt;
  __syncthreads();
  for (int s = 128; s > 0; s >>= 1) {
    if (tid < s) { ssum[tid] += ssum[tid + s]; scnt[tid] += scnt[tid + s]; }
    __syncthreads();
  }
  if (tid == 0) out[0] = (scnt[0] > 0) ? (ssum[0] / (float)scnt[0]) : 0.0f;
}

// ---------------------------------------------------------------------------
extern "C" void kernel_launch(void* const* d_in, const int* in_sizes, int n_in,
                              void* d_out, int out_size, void* d_ws, size_t ws_size,
                              hipStream_t stream) {
  const float* X      = (const float*)d_in[0];   // [4096, 512] f32
  const int*   labels = (const int*)d_in[1];     // [4096] int
  float* out = (float*)d_out;

  // Workspace: hpBits[4096] | hnBits[4096] | norms[4096]  (48 KB)
  int*   hpBits = (int*)d_ws;
  int*   hnBits = hpBits + B_N;
  float* norms  = (float*)(hnBits + B_N);

  norms_init_kernel<<<B_N, 32, 0, stream>>>(X, norms, hpBits, hnBits);
  tile_kernel<<<B_N / 16, 256, 0, stream>>>(X, labels, norms, hpBits, hnBits);
  finalize_kernel<<<1, 256, 0, stream>>>(hpBits, hnBits, out);
}